// MyConv2d_72189810311232
// MI455X (gfx1250) — compile-verified
//
#include <hip/hip_runtime.h>

typedef __attribute__((ext_vector_type(2))) float v2f;
typedef __attribute__((ext_vector_type(8))) float v8f;
typedef __attribute__((ext_vector_type(4))) unsigned int v4u;
typedef __attribute__((ext_vector_type(8))) int v8i;
typedef __attribute__((ext_vector_type(4))) int v4i;

#define IN_CH 32
#define OUT_CH 32
#define HW 256
#define OHW 254
#define NTHREADS 512

#define W_FLOATS (9 * 8 * 2 * 32 * 2)   // 9216 floats = 36864 B (A-fragments)
#define X_FLOATS (IN_CH * 3 * HW)       // 24576 floats = 98304 B (input strip)

// LDS weight layout: [tap(9)][kstep(8)][mtile(2)][lane(32)][2 floats]
// A-fragment for V_WMMA_F32_16X16X4_F32:
//   lanes 0-15 : M=lane,    V0=K0, V1=K1
//   lanes 16-31: M=lane-16, V0=K2, V1=K3
__global__ __launch_bounds__(NTHREADS) void conv3x3_wmma_f32(
    const float* __restrict__ x, const float* __restrict__ w,
    float* __restrict__ out)
{
    // Single LDS block, manually partitioned: weights at offset 0 so every
    // A-fragment ds_load_b64 uses a 16-bit immediate offset (<= 36608 B).
    __shared__ __attribute__((aligned(16))) float smem[W_FLOATS + X_FLOATS];
    float* lds_w = smem;              // [0      .. 36863] bytes
    float* lds_x = smem + W_FLOATS;   // [36864  .. 135167] bytes

    const int oh = blockIdx.x % OHW;
    const int n  = blockIdx.x / OHW;

    // ---- TDM: DMA the 3-row x 32-channel input strip into LDS ----
#if __has_builtin(__builtin_amdgcn_tensor_load_to_lds)
    if (threadIdx.x < 32) {   // one wave issues the tensor DMA
        unsigned long long ga = (unsigned long long)(size_t)(const void*)
            (x + (size_t)n * IN_CH * HW * HW + (size_t)oh * HW);
        unsigned int ldsa = (unsigned int)(size_t)(void*)lds_x;  // LDS byte offset
        // D# group 0: count=1 valid descriptor | lds_addr | global_addr | type=2
        v4u g0 = { 1u,
                   ldsa,
                   (unsigned int)(ga & 0xffffffffu),
                   (unsigned int)((ga >> 32) & 0x01ffffffu) | (2u << 30) };
        // D# group 1: data_size=4B; tensor 256x256 (xy), strides 256 / 65536;
        //             tile 256(x) x 3(y) x 32(z=ci)
        v8i g1 = { (int)(2u << 16),           // data_size = 2 -> 4 bytes
                   (int)((unsigned)HW << 16), // tensor_dim0[15:0]=256 (bits 63:48)
                   (int)((unsigned)HW << 16), // tensor_dim0 hi=0 | tensor_dim1[15:0]=256
                   (int)((unsigned)HW << 16), // tensor_dim1 hi=0 | tile_dim0=256
                   (int)(3u | (32u << 16)),   // tile_dim1=3 | tile_dim2=32
                   (int)HW,                   // tensor_dim0_stride = 256
                   0,                         // d0_stride hi | d1_stride[15:0]=0
                   1 };                       // tensor_dim1_stride[47:16]: 65536>>16
        // D# group 2: tensor_dim2=32, tensor_dim3=1, dim2_stride=0, tile_dim3=0
        v4i g2 = { 32, 1, 0, 0 };
        v4i g3 = { 0, 0, 0, 0 };
        v8i gz = { 0, 0, 0, 0, 0, 0, 0, 0 };  // extra group (clang-23 6-arg form)
        __builtin_amdgcn_tensor_load_to_lds(g0, g1, g2, g3, gz, 0);
        __builtin_amdgcn_s_wait_tensorcnt((short)0);
    }
#else
    // Fallback: cooperative copy (keeps kernel functional if builtin missing)
    {
        const float* xsrc = x + (size_t)n * IN_CH * HW * HW + (size_t)oh * HW;
        for (int i = threadIdx.x; i < X_FLOATS; i += NTHREADS) {
            int ci = i / (3 * HW);
            int r  = (i / HW) % 3;
            int c  = i % HW;
            lds_x[i] = xsrc[(size_t)ci * HW * HW + r * HW + c];
        }
    }
#endif

    // ---- stage weights into A-fragment layout ----
    for (int idx = threadIdx.x; idx < 9 * 8 * 2 * 32; idx += NTHREADS) {
        int L = idx & 31;
        int m = (idx >> 5) & 1;
        int s = (idx >> 6) & 7;
        int t = idx >> 9;                 // tap 0..8
        int kh = t / 3, kw = t % 3;
        int oc = m * 16 + (L & 15);
        int ci = s * 4 + (L >> 4) * 2;
        int g  = ((oc * IN_CH + ci) * 3 + kh) * 3 + kw;
        lds_w[idx * 2 + 0] = w[g];        // ci
        lds_w[idx * 2 + 1] = w[g + 9];    // ci+1 (stride K*K = 9)
    }
    __syncthreads();

    // ---- one wave -> 32(oc) x 16(pixel) tile; 16 waves cover the row ----
    int lane = threadIdx.x & 31;
    int tx   = threadIdx.x >> 5;          // pixel tile 0..15
    int px   = lane & 15;
    int half = lane >> 4;
    int ow   = tx * 16 + px;

    v8f c0 = {};   // oc 0..15
    v8f c1 = {};   // oc 16..31

    #pragma unroll
    for (int kh = 0; kh < 3; ++kh) {
        #pragma unroll
        for (int kw = 0; kw < 3; ++kw) {
            int iw = ow + kw;
            if (iw > HW - 1) iw = HW - 1;          // clamp (masked at store)
            int t = kh * 3 + kw;
            int bbase = half * 2 * (3 * HW) + kh * HW + iw;  // + ci*(3*HW)
            #pragma unroll
            for (int s = 0; s < 8; ++s) {
                v2f b;
                b.x = lds_x[bbase + (s * 4    ) * (3 * HW)];
                b.y = lds_x[bbase + (s * 4 + 1) * (3 * HW)];
                const v2f* af =
                    (const v2f*)&lds_w[(((t * 8 + s) * 2 + 0) * 32 + lane) * 2];
                v2f a0 = af[0];    // mtile 0
                v2f a1 = af[32];   // mtile 1
                c0 = __builtin_amdgcn_wmma_f32_16x16x4_f32(
                        false, a0, false, b, (short)0, c0, false, false);
                c1 = __builtin_amdgcn_wmma_f32_16x16x4_f32(
                        false, a1, false, b, (short)0, c1, false, false);
            }
        }
    }

    // ---- store: C/D layout lanes 0-15 -> M=r, lanes 16-31 -> M=8+r ----
    if (ow < OHW) {
        int ocb = half * 8;
        size_t obase = ((size_t)n * OUT_CH) * OHW * OHW + (size_t)oh * OHW + ow;
        #pragma unroll
        for (int r = 0; r < 8; ++r) {
            out[obase + (size_t)(ocb + r)      * OHW * OHW] = c0[r];
            out[obase + (size_t)(16 + ocb + r) * OHW * OHW] = c1[r];
        }
    }
}

extern "C" void kernel_launch(void* const* d_in, const int* in_sizes, int n_in,
                              void* d_out, int out_size, void* d_ws, size_t ws_size,
                              hipStream_t stream) {
    const float* x = (const float*)d_in[0];   // (16, 32, 256, 256) fp32
    const float* w = (const float*)d_in[1];   // (32, 32, 3, 3)     fp32
    float* out = (float*)d_out;               // (16, 32, 254, 254) fp32

    dim3 grid(16 * OHW);                      // one block per (n, oh): 4064
    conv3x3_wmma_f32<<<grid, NTHREADS, 0, stream>>>(x, w, out);
}